// DistillationLossWithChunk_5858335392431
// MI455X (gfx1250) — compile-verified
//
#include <hip/hip_runtime.h>

// DistillationLossWithChunk for MI455X (gfx1250).
// Memory-bound streaming reduction: single pass over student+teacher logits
// (1.05 GB -> ~45us at 23.3 TB/s HBM). Online-softmax accumulators, wave32
// shuffles, deterministic two-kernel reduction (no float atomics).
// Final reduction uses V_WMMA_F32_16X16X4_F32 (A=ones) as a full-precision
// 32-lane fp32 summation unit.

#define ALPHA_ 0.7f
#define TEMP_ 5.0f
#define PAD_ID_ 0
#define NUM_CHUNKS_ 4
#define BATCH_ 2
#define THREADS_ 256

typedef __attribute__((ext_vector_type(2))) float v2f;
typedef __attribute__((ext_vector_type(4))) float v4f;   // native vector: OK for nontemporal builtin
typedef __attribute__((ext_vector_type(8))) float v8f;

#define NEG_BIG (-3.0e38f)

// ---- online logsumexp helpers (rare-rescale form: ~1 exp per element) ----
__device__ __forceinline__ void online_mz(float& m, float& z, float a) {
    if (a <= m) {
        z += __expf(a - m);
    } else {
        z = z * __expf(m - a) + 1.0f;   // exp(m - a) underflows to 0 for sentinel m
        m = a;
    }
}

__device__ __forceinline__ void combine_mz(float& m, float& z, float m2, float z2) {
    float M = fmaxf(m, m2);
    z = z * __expf(m - M) + z2 * __expf(m2 - M);
    m = M;
}

__device__ __forceinline__ void combine_mzw(float& m, float& z, float& w,
                                            float m2, float z2, float w2) {
    float M  = fmaxf(m, m2);
    float e1 = __expf(m - M);
    float e2 = __expf(m2 - M);
    z = z * e1 + z2 * e2;
    w = w * e1 + w2 * e2;
    m = M;
}

__device__ __forceinline__ void wave_reduce_mz(float& m, float& z) {
    #pragma unroll
    for (int off = 16; off > 0; off >>= 1) {
        float m2 = __shfl_xor(m, off, 32);
        float z2 = __shfl_xor(z, off, 32);
        combine_mz(m, z, m2, z2);
    }
}

__device__ __forceinline__ void wave_reduce_mzw(float& m, float& z, float& w) {
    #pragma unroll
    for (int off = 16; off > 0; off >>= 1) {
        float m2 = __shfl_xor(m, off, 32);
        float z2 = __shfl_xor(z, off, 32);
        float w2 = __shfl_xor(w, off, 32);
        combine_mzw(m, z, w, m2, z2, w2);
    }
}

// ---- kernel 1: one block per token; stream the 32000-wide row once ----
__global__ __launch_bounds__(THREADS_) void distill_token_kernel(
    const float* __restrict__ s_logits,
    const float* __restrict__ t_logits,
    const int*   __restrict__ labels,
    float* __restrict__ ws_kl,
    float* __restrict__ ws_nll,
    float* __restrict__ ws_valid,
    int V) {

    const int token  = blockIdx.x;
    const int tid    = threadIdx.x;
    const int lane   = tid & 31;
    const int wave   = tid >> 5;
    const int chunkV  = V / NUM_CHUNKS_;
    const int chunkV4 = chunkV >> 2;                  // 4-float groups per chunk
    const float invT = 1.0f / TEMP_;

    const v4f* srow = (const v4f*)(s_logits + (size_t)token * V);
    const v4f* trow = (const v4f*)(t_logits + (size_t)token * V);

    __shared__ float sm_t[8], sz_t[8], sw_t[8];
    __shared__ float sm_s[8], sz_s[8];

    float m_f = NEG_BIG, z_f = 0.0f;   // full-vocab student lse (T=1), across chunks
    float kl_acc = 0.0f;               // meaningful on tid==0 only

    for (int c = 0; c < NUM_CHUNKS_; ++c) {
        float m_s = NEG_BIG, z_s = 0.0f;
        float m_t = NEG_BIG, z_t = 0.0f, w_t = 0.0f;

        const v4f* sc = srow + (size_t)c * chunkV4;
        const v4f* tc = trow + (size_t)c * chunkV4;

        for (int i = tid; i < chunkV4; i += THREADS_) {
            v4f s4 = __builtin_nontemporal_load(&sc[i]);   // streaming: NT hint, 1 GB >> L2
            v4f t4 = __builtin_nontemporal_load(&tc[i]);
            #pragma unroll
            for (int k = 0; k < 4; ++k) {
                float s = s4[k];
                float t = t4[k];
                // student chunk lse @ T
                online_mz(m_s, z_s, s * invT);
                // student full-vocab lse @ 1
                online_mz(m_f, z_f, s);
                // teacher chunk lse @ T, fused with W = sum e_t * (t-s)/T
                float b = t * invT;
                float d = (t - s) * invT;
                if (b <= m_t) {
                    float e = __expf(b - m_t);
                    z_t += e;
                    w_t += e * d;
                } else {
                    float r = __expf(m_t - b);
                    z_t = z_t * r + 1.0f;
                    w_t = w_t * r + d;
                    m_t = b;
                }
            }
        }

        wave_reduce_mz(m_s, z_s);
        wave_reduce_mzw(m_t, z_t, w_t);
        if (lane == 0) {
            sm_t[wave] = m_t; sz_t[wave] = z_t; sw_t[wave] = w_t;
            sm_s[wave] = m_s; sz_s[wave] = z_s;
        }
        __syncthreads();
        if (tid == 0) {
            float M = sm_t[0], Z = sz_t[0], W = sw_t[0];
            float Ms = sm_s[0], Zs = sz_s[0];
            #pragma unroll
            for (int j = 1; j < 8; ++j) {
                combine_mzw(M, Z, W, sm_t[j], sz_t[j], sw_t[j]);
                combine_mz(Ms, Zs, sm_s[j], sz_s[j]);
            }
            // sum p_t*(log_t - log_s) = W/Z - lse_t + lse_s
            kl_acc += W / Z - (M + logf(Z)) + (Ms + logf(Zs));
        }
        __syncthreads();   // protect shared arrays before next chunk
    }

    // full-vocab student lse -> CE term
    wave_reduce_mz(m_f, z_f);
    if (lane == 0) { sm_s[wave] = m_f; sz_s[wave] = z_f; }
    __syncthreads();
    if (tid == 0) {
        float M = sm_s[0], Z = sz_s[0];
        #pragma unroll
        for (int j = 1; j < 8; ++j) combine_mz(M, Z, sm_s[j], sz_s[j]);
        float lse_full = M + logf(Z);

        int lab = labels[token];
        float s_lab = s_logits[(size_t)token * V + lab];
        bool valid = (lab != PAD_ID_);

        ws_kl[token]    = kl_acc;
        ws_nll[token]   = valid ? (lse_full - s_lab) : 0.0f;
        ws_valid[token] = valid ? 1.0f : 0.0f;
    }
}

// ---- WMMA-based fp32 wave sum: D = ones(16x4) x B(4x16) + 0 ----
// Each lane contributes one value in B's first VGPR (second VGPR = 0).
// Every row of D holds the 16 column sums; summing row 0 (lanes 0..15 of d[0],
// via xor-shuffles that stay inside the 16-lane half) yields the sum of all
// 32 lane values exactly (full fp32, independent of the (k,n) lane mapping).
__device__ __forceinline__ float wave_sum_wmma(float x) {
    v2f a; a.x = 1.0f; a.y = 1.0f;
    v2f b; b.x = x;    b.y = 0.0f;
    v8f c = {};
    v8f d = __builtin_amdgcn_wmma_f32_16x16x4_f32(
        /*neg_a=*/false, a, /*neg_b=*/false, b,
        /*c_mod=*/(short)0, c, /*reuse_a=*/false, /*reuse_b=*/false);
    float r = d[0];
    #pragma unroll
    for (int off = 8; off > 0; off >>= 1) r += __shfl_xor(r, off, 32);
    return r;   // lane 0 (and lane 16) hold the exact wave sum
}

// ---- kernel 2: deterministic final reduction over tokens ----
__global__ __launch_bounds__(THREADS_) void distill_finalize_kernel(
    const float* __restrict__ ws_kl,
    const float* __restrict__ ws_nll,
    const float* __restrict__ ws_valid,
    float* __restrict__ out,
    int n_tokens) {

    const int tid  = threadIdx.x;
    const int lane = tid & 31;
    const int wave = tid >> 5;

    float kl = 0.0f, nll = 0.0f, val = 0.0f;
    for (int i = tid; i < n_tokens; i += THREADS_) {
        kl  += ws_kl[i];
        nll += ws_nll[i];
        val += ws_valid[i];
    }

    // EXEC is all-ones here (uniform control flow) -> WMMA legal.
    float kl_w  = wave_sum_wmma(kl);
    float nll_w = wave_sum_wmma(nll);
    float val_w = wave_sum_wmma(val);

    __shared__ float skl[8], snll[8], sval[8];
    if (lane == 0) { skl[wave] = kl_w; snll[wave] = nll_w; sval[wave] = val_w; }
    __syncthreads();
    if (tid == 0) {
        float K = 0.0f, N = 0.0f, Vv = 0.0f;
        #pragma unroll
        for (int j = 0; j < 8; ++j) { K += skl[j]; N += snll[j]; Vv += sval[j]; }
        float total_kl = K * (TEMP_ * TEMP_) / (float)(BATCH_ * NUM_CHUNKS_);
        float n_valid  = (Vv < 1.0f) ? 1.0f : Vv;
        float ce = N / n_valid;
        out[0] = ALPHA_ * total_kl + (1.0f - ALPHA_) * ce;
    }
}

extern "C" void kernel_launch(void* const* d_in, const int* in_sizes, int n_in,
                              void* d_out, int out_size, void* d_ws, size_t ws_size,
                              hipStream_t stream) {
    const float* s_logits = (const float*)d_in[0];
    const float* t_logits = (const float*)d_in[1];
    const int*   labels   = (const int*)d_in[2];   // harness contract: integer -> int*

    const int n_tokens = in_sizes[2];              // B*S = 4096
    const int V        = in_sizes[0] / n_tokens;   // 32000

    float* ws_kl    = (float*)d_ws;
    float* ws_nll   = ws_kl + n_tokens;
    float* ws_valid = ws_nll + n_tokens;

    distill_token_kernel<<<n_tokens, THREADS_, 0, stream>>>(
        s_logits, t_logits, labels, ws_kl, ws_nll, ws_valid, V);
    distill_finalize_kernel<<<1, THREADS_, 0, stream>>>(
        ws_kl, ws_nll, ws_valid, (float*)d_out, n_tokens);
}